// MLP_Direct_23390391894488
// MI455X (gfx1250) — compile-verified
//
#include <hip/hip_runtime.h>

typedef __attribute__((ext_vector_type(2))) float v2f;
typedef __attribute__((ext_vector_type(8))) float v8f;

#define TILE_B   16
#define NCOL     800
#define KDIM     784
#define WAVES    10
#define TPB      (WAVES * 32)
#define CPW      5            // 16-col tiles per wave: 10 waves * 5 * 16 = 800
#define XS_STRIDE 788         // 788 % 64 = 20 -> 16 rows map to 16 distinct 4-bank groups
#define CT_STRIDE 804         // 804 % 64 = 36 -> conflict-free across 16 rows

__global__ __launch_bounds__(TPB)
void snn_fused(const float* __restrict__ inp,
               const float* __restrict__ W1,
               const float* __restrict__ W2,
               float* __restrict__ out)
{
    // Reused buffer: phase 1 = x tile [16][788], phase 2 = spike counts [16][804]
    __shared__ float lds[16 * CT_STRIDE];

    const int tid  = threadIdx.x;
    const int lane = tid & 31;
    const int wid  = tid >> 5;
    const int m    = lane & 15;     // A-row (batch) / B-col index within a 16-tile
    const int hi   = lane >> 4;     // 0: K=0,1  1: K=2,3
    const int koff = hi * 2;
    const int rowbase = blockIdx.x * TILE_B;

    // ---- Stage x tile [16, 784] into LDS (float4, coalesced) ----
    {
        const float4* src = (const float4*)(inp + (size_t)rowbase * KDIM);
        for (int idx = tid; idx < 16 * (KDIM / 4); idx += TPB) {
            int r  = idx / (KDIM / 4);
            int c4 = idx - r * (KDIM / 4);
            float4 v = src[r * (KDIM / 4) + c4];
            *(float4*)(&lds[r * XS_STRIDE + c4 * 4]) = v;
        }
    }
    __syncthreads();

    // ---- GEMM1: static = x @ W1^T, fp32 WMMA 16x16x4, 5 col-tiles per wave ----
    v8f acc[CPW] = {};
    const float* pB[CPW];
#pragma unroll
    for (int t = 0; t < CPW; ++t) {
        int n = (wid * CPW + t) * 16 + m;          // W1 row = output column
        pB[t] = W1 + (size_t)n * KDIM + koff;
    }
    const int aBase = m * XS_STRIDE + koff;

    for (int k = 0; k < KDIM; k += 4) {
        v2f a = *(const v2f*)(&lds[aBase + k]);    // A: 16x4 fp32 fragment from LDS
#pragma unroll
        for (int t = 0; t < CPW; ++t) {
            v2f b = *(const v2f*)(pB[t] + k);      // B: 4x16 fp32 fragment (W1^T)
            acc[t] = __builtin_amdgcn_wmma_f32_16x16x4_f32(
                false, a, false, b, (short)0, acc[t], false, false);
        }
    }

    // ---- 32-step LIF dynamics, registers only; accumulate spike counts ----
    float mem[CPW][8];
    float cnt[CPW][8];
#pragma unroll
    for (int t = 0; t < CPW; ++t)
#pragma unroll
        for (int i = 0; i < 8; ++i) { mem[t][i] = 0.f; cnt[t][i] = 0.f; }

    for (int step = 0; step < 32; ++step) {
#pragma unroll
        for (int t = 0; t < CPW; ++t)
#pragma unroll
            for (int i = 0; i < 8; ++i) {
                float mm  = 0.95f * mem[t][i] + 0.05f * acc[t][i];
                float thr = mm - 1.0f;
                bool  s   = thr > 0.0f;
                cnt[t][i] += s ? 1.0f : 0.0f;
                mem[t][i]  = s ? (mm - 1.0f) : mm;
            }
    }

    // ---- Scatter counts to LDS: C[16][800] (stride 804) ----
    __syncthreads();   // all waves done reading x tile; safe to overwrite
#pragma unroll
    for (int t = 0; t < CPW; ++t)
#pragma unroll
        for (int i = 0; i < 8; ++i) {
            int row = i + hi * 8;                       // C/D layout: VGPR i -> M=i (+8 for hi lanes)
            int col = (wid * CPW + t) * 16 + m;
            lds[row * CT_STRIDE + col] = cnt[t][i];
        }
    __syncthreads();

    // ---- GEMM2 (wave 0): out = (C @ W2^T) / 32, fp32 WMMA, K = 800 ----
    if (wid == 0) {
        v8f c2 = {};
        int nc = (m < 10) ? m : 0;                      // lanes 10..15 alias row 0; never stored
        const float* pB2 = W2 + (size_t)nc * NCOL + koff;
        const int a2Base = m * CT_STRIDE + koff;

        for (int k = 0; k < NCOL; k += 4) {
            v2f a = *(const v2f*)(&lds[a2Base + k]);
            v2f b = *(const v2f*)(pB2 + k);
            c2 = __builtin_amdgcn_wmma_f32_16x16x4_f32(
                false, a, false, b, (short)0, c2, false, false);
        }

        if (m < 10) {
#pragma unroll
            for (int i = 0; i < 8; ++i) {
                int r = i + hi * 8;
                out[(size_t)(rowbase + r) * 10 + m] = c2[i] * 0.03125f;
            }
        }
    }
}

extern "C" void kernel_launch(void* const* d_in, const int* in_sizes, int n_in,
                              void* d_out, int out_size, void* d_ws, size_t ws_size,
                              hipStream_t stream) {
    const float* inp = (const float*)d_in[0];   // [16384, 1, 28, 28] -> [B, 784]
    const float* W1  = (const float*)d_in[1];   // [800, 784]
    const float* W2  = (const float*)d_in[2];   // [10, 800]
    float* out = (float*)d_out;                 // [B, 10]

    int B = in_sizes[0] / KDIM;                 // 16384
    dim3 grid(B / TILE_B);
    snn_fused<<<grid, TPB, 0, stream>>>(inp, W1, W2, out);
}